// PNATower_82489141887071
// MI455X (gfx1250) — compile-verified
//
#include <hip/hip_runtime.h>
#include <hip/hip_bf16.h>
#include <math.h>

#define NNODES 100000
#define NEDGES 1200000
#define DIN    64
#define DOUT   64
#define EDIM   16
#define KPRE   144              // 2*DIN + EDIM
#define KPOST  832              // (4*3+1)*DIN
#define AVG_D  2.5649493574615367f
#define EPSV   1e-05f

typedef __attribute__((ext_vector_type(2))) float v2f;
typedef __attribute__((ext_vector_type(8))) float v8f;

// ---- order-preserving float<->uint encode for atomic max/min ----
__device__ __forceinline__ unsigned fenc(float f) {
    unsigned u = __float_as_uint(f);
    return (u & 0x80000000u) ? ~u : (u | 0x80000000u);
}
__device__ __forceinline__ float fdec(unsigned u) {
    return (u & 0x80000000u) ? __uint_as_float(u & 0x7fffffffu)
                             : __uint_as_float(~u);
}

// ---------------- workspace layout (floats) ----------------
// s1   : N*64    segment sum of m
// s2   : N*64    segment sum of m*m
// mx   : N*64    encoded segment max
// mn   : N*64    encoded segment min
// deg  : N       degree (float)
// bns  : 64      BN column sum
// bnq  : 64      BN column sum-of-squares

// ================= init =================
__global__ __launch_bounds__(256) void pna_init(float* s1, float* s2,
                                                unsigned* mx, unsigned* mn,
                                                float* deg, float* bns, float* bnq) {
    int i = blockIdx.x * 256 + threadIdx.x;
    if (i < NNODES * DIN) {
        s1[i] = 0.0f; s2[i] = 0.0f;
        mx[i] = 0u;            // below every encoded real (deg>=1 guarantees overwrite)
        mn[i] = 0xFFFFFFFFu;   // above every encoded real
    }
    if (i < NNODES) deg[i] = 0.0f;
    if (i < DOUT) { bns[i] = 0.0f; bnq[i] = 0.0f; }
}

// ================= degree =================
__global__ __launch_bounds__(256) void pna_degree(const int* dst, float* deg) {
    int i = blockIdx.x * 256 + threadIdx.x;
    if (i < NEDGES) atomicAdd(&deg[dst[i]], 1.0f);
}

// ================= edge GEMM + fused segment aggregation =================
// block = 256 threads (8 waves), 32 edges per block. E/32 = 37500 blocks exactly.
#define ZSTRIDE 148   // 144 padded; 148 mod 64 = 20 -> conflict-free A reads
__global__ __launch_bounds__(256) void pna_edge(
    const float* __restrict__ h, const float* __restrict__ e,
    const float* __restrict__ Wpre, const float* __restrict__ bpre,
    const int* __restrict__ src, const int* __restrict__ dst,
    float* s1, float* s2, unsigned* mx, unsigned* mn) {

    __shared__ float zs[32 * ZSTRIDE];        // gathered z rows
    __shared__ float wsT[DIN * ZSTRIDE];      // W_pre transposed: wsT[n][k]
    __shared__ int   ldst[32];

    const int tid = threadIdx.x;
    const int e0  = blockIdx.x * 32;

    // ---- stage W_pre^T (coalesced global read) ----
    for (int i = tid; i < KPRE * DIN; i += 256) {
        int n = i & 63, k = i >> 6;           // i = k*64 + n
        wsT[n * ZSTRIDE + k] = Wpre[k * DIN + n];
    }
    // ---- gather z = [h[src], h[dst], e] : 8 threads per edge ----
    {
        int r = tid >> 3, p = tid & 7;        // edge row 0..31, part 0..7
        int ei = e0 + r;
        int sI = src[ei], dI = dst[ei];
        if (p == 0) ldst[r] = dI;
        int c0 = p * 18;
        for (int c = c0; c < c0 + 18; ++c) {
            float v;
            if (c < DIN)            v = h[sI * DIN + c];
            else if (c < 2 * DIN)   v = h[dI * DIN + (c - DIN)];
            else                    v = e[ei * EDIM + (c - 2 * DIN)];
            zs[r * ZSTRIDE + c] = v;
        }
    }
    __syncthreads();

    // ---- WMMA: wave -> 16x16 tile; eg selects edge half, cg selects col group ----
    const int l  = tid & 31;
    const int wv = tid >> 5;
    const int cg = wv & 3;        // column group: 16 cols
    const int eg = wv >> 2;       // edge group: 16 edges
    const int lo = l & 15;
    const int hi = l >> 4;

    v8f acc = {};
    const float* arow = &zs[(eg * 16 + lo) * ZSTRIDE + 2 * hi];
    const float* brow = &wsT[(cg * 16 + lo) * ZSTRIDE + 2 * hi];
#pragma unroll 4
    for (int k0 = 0; k0 < KPRE; k0 += 4) {
        v2f a = *(const v2f*)(arow + k0);
        v2f b = *(const v2f*)(brow + k0);
        acc = __builtin_amdgcn_wmma_f32_16x16x4_f32(
            false, a, false, b, (short)0, acc, false, false);
    }

    // ---- epilogue: bias + atomic segment aggregation ----
    const int col = cg * 16 + lo;
    const float bv = bpre[col];
#pragma unroll
    for (int r = 0; r < 8; ++r) {
        int M  = r + 8 * hi;              // row within 16-edge tile
        int er = eg * 16 + M;
        int d  = ldst[er];
        float v = acc[r] + bv;
        unsigned ev = fenc(v);
        size_t o = (size_t)d * DIN + col;
        atomicAdd(&s1[o], v);
        atomicAdd(&s2[o], v * v);
        atomicMax(&mx[o], ev);
        atomicMin(&mn[o], ev);
    }
}

// ================= node feature build + GEMM + BN partials =================
// block = 128 threads (4 waves), 16 nodes per block. N/16 = 6250 blocks exactly.
#define FSTRIDE 836   // 832 padded (+4 extra; slot 832 holds logd). 836 mod 64 = 4
__global__ __launch_bounds__(128) void pna_node(
    const float* __restrict__ h, const float* __restrict__ snorm,
    const float* __restrict__ Wpost, const float* __restrict__ bpost,
    const float* __restrict__ s1, const float* __restrict__ s2,
    const unsigned* __restrict__ mx, const unsigned* __restrict__ mn,
    const float* __restrict__ deg,
    float* __restrict__ out, float* bns, float* bnq) {

    __shared__ float feat[16 * FSTRIDE];
    const int tid = threadIdx.x;
    const int n0  = blockIdx.x * 16;

    // ---- phase 1: h + agg = [h, mean, max, min, std], plus logd per row ----
    for (int i = tid; i < 16 * 320; i += 128) {
        int r = i / 320, c = i % 320;
        int node = n0 + r;
        float v;
        if (c < DIN) {
            v = h[node * DIN + c];
        } else {
            int a = c - DIN;
            int sub = a >> 6, colc = a & 63;
            size_t o = (size_t)node * DIN + colc;
            float D  = deg[node];
            if (sub == 0)      v = s1[o] / D;
            else if (sub == 1) v = fdec(mx[o]);
            else if (sub == 2) v = fdec(mn[o]);
            else {
                float mean = s1[o] / D;
                v = sqrtf(fmaxf(s2[o] / D - mean * mean, 0.0f) + EPSV);
            }
            if (c == DIN) feat[r * FSTRIDE + 832] = logf(D + 1.0f);
        }
        feat[r * FSTRIDE + c] = v;
    }
    __syncthreads();
    // ---- phase 2: scaled copies agg*(logd/AVG), agg*(AVG/logd) ----
    for (int i = tid; i < 16 * 512; i += 128) {
        int r = i >> 9, j = i & 511;
        float logd = feat[r * FSTRIDE + 832];
        float f = (j < 256) ? (logd / AVG_D) : (AVG_D / logd);
        feat[r * FSTRIDE + 320 + j] = feat[r * FSTRIDE + DIN + (j & 255)] * f;
    }
    __syncthreads();

    // ---- WMMA: 4 waves, each 16 nodes x 16 cols, K = 832 ----
    const int l  = tid & 31;
    const int cg = tid >> 5;
    const int lo = l & 15;
    const int hi = l >> 4;
    const int col = cg * 16 + lo;

    v8f acc = {};
    const float* arow = &feat[lo * FSTRIDE + 2 * hi];
#pragma unroll 4
    for (int k0 = 0; k0 < KPOST; k0 += 4) {
        v2f a = *(const v2f*)(arow + k0);
        v2f b;
        b.x = Wpost[(k0 + 2 * hi)     * DOUT + col];
        b.y = Wpost[(k0 + 2 * hi + 1) * DOUT + col];
        acc = __builtin_amdgcn_wmma_f32_16x16x4_f32(
            false, a, false, b, (short)0, acc, false, false);
    }

    // ---- epilogue: bias, snorm, store pre-BN out, BN partial sums ----
    const float bv = bpost[col];
    float ps = 0.0f, ps2 = 0.0f;
#pragma unroll
    for (int r = 0; r < 8; ++r) {
        int M = r + 8 * hi;
        int node = n0 + M;
        float v = (acc[r] + bv) * snorm[node];
        out[(size_t)node * DOUT + col] = v;
        ps += v; ps2 += v * v;
    }
    // combine the two half-wave rows (wave32), then one atomic per column
    ps  += __shfl_xor(ps, 16, 32);
    ps2 += __shfl_xor(ps2, 16, 32);
    if (hi == 0) {
        atomicAdd(&bns[col], ps);
        atomicAdd(&bnq[col], ps2);
    }
}

// ================= batch-norm finalize =================
__global__ __launch_bounds__(256) void pna_bn(
    float* out, const float* bns, const float* bnq,
    const float* gamma, const float* beta) {
    int i = blockIdx.x * 256 + threadIdx.x;
    if (i >= NNODES * DOUT) return;
    int c = i & 63;
    float mu  = bns[c] / (float)NNODES;
    float var = bnq[c] / (float)NNODES - mu * mu;
    float v = out[i];
    out[i] = (v - mu) * rsqrtf(var + 1e-05f) * gamma[c] + beta[c];
}

extern "C" void kernel_launch(void* const* d_in, const int* in_sizes, int n_in,
                              void* d_out, int out_size, void* d_ws, size_t ws_size,
                              hipStream_t stream) {
    const float* h      = (const float*)d_in[0];
    const float* e      = (const float*)d_in[1];
    const float* snorm  = (const float*)d_in[2];
    const float* Wpre   = (const float*)d_in[3];
    const float* bpre   = (const float*)d_in[4];
    const float* Wpost  = (const float*)d_in[5];
    const float* bpost  = (const float*)d_in[6];
    const float* gamma  = (const float*)d_in[7];
    const float* beta   = (const float*)d_in[8];
    const int*   src    = (const int*)d_in[9];
    const int*   dst    = (const int*)d_in[10];
    float* out = (float*)d_out;

    float*    s1  = (float*)d_ws;
    float*    s2  = s1 + (size_t)NNODES * DIN;
    unsigned* mx  = (unsigned*)(s2 + (size_t)NNODES * DIN);
    unsigned* mn  = mx + (size_t)NNODES * DIN;
    float*    deg = (float*)(mn + (size_t)NNODES * DIN);
    float*    bns = deg + NNODES;
    float*    bnq = bns + DOUT;

    pna_init  <<<(NNODES * DIN + 255) / 256, 256, 0, stream>>>(s1, s2, mx, mn, deg, bns, bnq);
    pna_degree<<<(NEDGES + 255) / 256, 256, 0, stream>>>(dst, deg);
    pna_edge  <<<NEDGES / 32, 256, 0, stream>>>(h, e, Wpre, bpre, src, dst, s1, s2, mx, mn);
    pna_node  <<<NNODES / 16, 128, 0, stream>>>(h, snorm, Wpost, bpost, s1, s2, mx, mn, deg,
                                                out, bns, bnq);
    pna_bn    <<<(NNODES * DOUT + 255) / 256, 256, 0, stream>>>(out, bns, bnq, gamma, beta);
}